// GCNLayer_58368605553168
// MI455X (gfx1250) — compile-verified
//
#include <hip/hip_runtime.h>

// GCN layer: mean-aggregate over in-edges, h = relu(agg @ W^T) + x
// N=100000 nodes, E=1600000 edges, D=128.
// d_out is reused as the msg_sum accumulator (same size), d_ws holds deg.

#define N_NODES 100000
#define N_EDGES 1600000
#define D 128
#define LDS_STRIDE 132   // 128 + 4-float pad -> conflict-free b64 LDS reads

typedef float v2f __attribute__((ext_vector_type(2)));
typedef float v8f __attribute__((ext_vector_type(8)));

// ---------------- Kernel A: zero msg accumulator (d_out) and deg ------------
__global__ void gcn_zero_kernel(float4* __restrict__ msg4, float* __restrict__ deg) {
    int tid = blockIdx.x * blockDim.x + threadIdx.x;   // 3.2M threads
    if (tid < N_NODES * (D / 4)) {
        msg4[tid] = make_float4(0.f, 0.f, 0.f, 0.f);
    }
    if (tid < N_NODES) deg[tid] = 0.f;
}

// ---------------- Kernel B: edge scatter (one wave32 per edge) --------------
// x is L2-resident (51.2 MB < 192 MB), so gathers hit L2; cost is dominated by
// the 204.8M f32 atomics, which resolve at the L2 atomic units.
__global__ __launch_bounds__(256) void gcn_scatter_kernel(
        const float4* __restrict__ x4,
        const int* __restrict__ src,
        const int* __restrict__ dst,
        float* __restrict__ msg,
        float* __restrict__ deg) {
    int wave = threadIdx.x >> 5;
    int lane = threadIdx.x & 31;
    int e = blockIdx.x * 8 + wave;
    if (e >= N_EDGES) return;
    int s = src[e];
    int d = dst[e];
    float4 v = x4[(size_t)s * (D / 4) + lane];          // 16B per lane, coalesced row
    float* mrow = msg + (size_t)d * D + lane * 4;
    atomicAdd(mrow + 0, v.x);
    atomicAdd(mrow + 1, v.y);
    atomicAdd(mrow + 2, v.z);
    atomicAdd(mrow + 3, v.w);
    if (lane == 0) atomicAdd(deg + d, 1.0f);
}

// ---------------- Kernel C: normalize + WMMA f32 GEMM + relu + residual -----
// One WG per 16-row tile. 8 waves each own a 16-col N-tile of the 128 outputs.
// h[m][n] = sum_k agg[m][k] * W[n][k]  (B[k][n] = W[n][k], W row-major [out][in])
__global__ __launch_bounds__(256) void gcn_gemm_kernel(
        const float* __restrict__ x,
        const float* __restrict__ W,
        const float* __restrict__ deg,
        float* __restrict__ out /* in: msg_sum, out: relu(agg@W^T)+x */) {

    __shared__ float ldsA[16 * LDS_STRIDE];

    const int row0 = blockIdx.x * 16;
    const int tid  = threadIdx.x;

    // Stage normalized agg tile (16 x 128) into LDS. deg==0 -> keep x.
    #pragma unroll
    for (int i = 0; i < 8; ++i) {
        int idx = tid + i * 256;          // 0..2047
        int r = idx >> 7;                 // row in tile
        int c = idx & 127;                // channel
        int g = (row0 + r) * D + c;
        float dg = deg[row0 + r];
        float m  = out[g];                // msg_sum accumulated by kernel B
        ldsA[r * LDS_STRIDE + c] = (dg > 0.f) ? (m / dg) : x[g];
    }
    __syncthreads();

    const int wave  = tid >> 5;           // 0..7 -> N-tile
    const int lane  = tid & 31;
    const int n0    = wave * 16;
    const int rcol  = lane & 15;          // A row / B column within tile
    const int khalf = (lane >> 4) * 2;    // K sub-offset: 0 (lanes 0-15) or 2

    // A-frag source: ldsA[rcol][k0 + khalf + {0,1}]
    const float* lrow = ldsA + rcol * LDS_STRIDE + khalf;
    // B-frag source: W[(n0+rcol)][k0 + khalf + {0,1}]  (64KB, L2/WGP$ hot)
    const float* wrow = W + (size_t)(n0 + rcol) * D + khalf;

    v8f c0 = {};
    v8f c1 = {};
    #pragma unroll
    for (int k0 = 0; k0 < D; k0 += 8) {   // two independent accumulator chains
        v2f a0 = *(const v2f*)(lrow + k0);
        v2f b0 = *(const v2f*)(wrow + k0);
        c0 = __builtin_amdgcn_wmma_f32_16x16x4_f32(
                 false, a0, false, b0, (short)0, c0, false, false);
        v2f a1 = *(const v2f*)(lrow + k0 + 4);
        v2f b1 = *(const v2f*)(wrow + k0 + 4);
        c1 = __builtin_amdgcn_wmma_f32_16x16x4_f32(
                 false, a1, false, b1, (short)0, c1, false, false);
    }

    // C/D layout: VGPR r -> M=r (lanes 0-15) or M=r+8 (lanes 16-31); N = lane%16.
    const int rbase = (lane < 16) ? 0 : 8;
    const int col   = n0 + rcol;
    #pragma unroll
    for (int r = 0; r < 8; ++r) {
        float h = c0[r] + c1[r];
        h = fmaxf(h, 0.f);
        int grow = row0 + rbase + r;
        out[(size_t)grow * D + col] = h + x[(size_t)grow * D + col];
    }
}

// ---------------------------------------------------------------------------
extern "C" void kernel_launch(void* const* d_in, const int* in_sizes, int n_in,
                              void* d_out, int out_size, void* d_ws, size_t ws_size,
                              hipStream_t stream) {
    const float* x   = (const float*)d_in[0];
    const int*   src = (const int*)d_in[1];   // int32 per harness contract
    const int*   dst = (const int*)d_in[2];
    const float* W   = (const float*)d_in[3];
    float* out = (float*)d_out;               // doubles as msg_sum accumulator
    float* deg = (float*)d_ws;                // 100000 floats (400 KB)

    // A: zero accumulators (3.2M threads: one float4 of msg each; first 100K also deg)
    gcn_zero_kernel<<<(N_NODES * (D / 4) + 255) / 256, 256, 0, stream>>>(
        (float4*)out, deg);

    // B: one wave per edge, 8 edges per 256-thread block
    gcn_scatter_kernel<<<N_EDGES / 8, 256, 0, stream>>>(
        (const float4*)x, src, dst, out, deg);

    // C: one WG per 16 node rows
    gcn_gemm_kernel<<<N_NODES / 16, 256, 0, stream>>>(x, W, deg, out);
}